// GINNodeRegressor_11888469475718
// MI455X (gfx1250) — compile-verified
//
#include <hip/hip_runtime.h>

#define GIN_D 64
#define TSTR 68   // LDS tile row stride (floats): TSTR*8 % 64 == 32 -> conflict-free halves
#define WSTR 68   // LDS transposed-weight row stride

typedef __attribute__((ext_vector_type(2))) float v2f;
typedef __attribute__((ext_vector_type(8))) float v8f;

// ---------------- zero workspace ----------------
__global__ __launch_bounds__(256) void gin_zero_kernel(float4* __restrict__ p, int n4) {
    int i = blockIdx.x * 256 + threadIdx.x;
    if (i < n4) p[i] = make_float4(0.f, 0.f, 0.f, 0.f);
}

// ---------------- edge aggregation: agg[dst] += h[src] ----------------
// 16 threads per edge, one float4 each -> 4 hardware f32 atomics per thread.
__global__ __launch_bounds__(256) void gin_agg_kernel(const float* __restrict__ h,
                                                      const int* __restrict__ src,
                                                      const int* __restrict__ dst,
                                                      float* __restrict__ agg, int e) {
    int tid  = blockIdx.x * 256 + threadIdx.x;
    int edge = tid >> 4;
    if (edge >= e) return;
    int c = (tid & 15) << 2;
    int s = src[edge];
    int d = dst[edge];
    const float4 v = *(const float4*)(h + (size_t)s * GIN_D + c);
    float* ap = agg + (size_t)d * GIN_D + c;
    unsafeAtomicAdd(ap + 0, v.x);
    unsafeAtomicAdd(ap + 1, v.y);
    unsafeAtomicAdd(ap + 2, v.z);
    unsafeAtomicAdd(ap + 3, v.w);
}

// ---------------- fused GIN MLP: hout = relu(relu((hin+agg)@W1+b1)@W2+b2) ----------------
// 128 threads = 4 waves, each wave owns a 16x64 row tile. FP32 WMMA 16x16x4.
__global__ __launch_bounds__(128)
void gin_mlp_kernel(const float* __restrict__ hin, const float* __restrict__ agg,
                    const float* __restrict__ w1, const float* __restrict__ b1,
                    const float* __restrict__ w2, const float* __restrict__ b2,
                    float* __restrict__ hout, int n) {
    __shared__ float sW1t[GIN_D * WSTR];   // transposed: sW1t[nCol*WSTR + k] = w1[k*D + nCol]
    __shared__ float sW2t[GIN_D * WSTR];
    __shared__ float sB1[GIN_D], sB2[GIN_D];
    __shared__ float sT[4][16 * TSTR];     // per-wave 16x64 staging tile (padded)

    const int t    = threadIdx.x;
    const int wave = t >> 5;
    const int lane = t & 31;
    const int m    = lane & 15;   // row (A) or col (B, C/D) within tile
    const int hi   = lane >> 4;   // half-wave select

    // Stage both weight matrices transposed into LDS (so B fragments are contiguous float2).
    for (int i = t; i < GIN_D * GIN_D / 4; i += 128) {
        int k  = i >> 4;
        int c4 = (i & 15) << 2;
        float4 a = ((const float4*)w1)[i];
        sW1t[(c4 + 0) * WSTR + k] = a.x;
        sW1t[(c4 + 1) * WSTR + k] = a.y;
        sW1t[(c4 + 2) * WSTR + k] = a.z;
        sW1t[(c4 + 3) * WSTR + k] = a.w;
        float4 b = ((const float4*)w2)[i];
        sW2t[(c4 + 0) * WSTR + k] = b.x;
        sW2t[(c4 + 1) * WSTR + k] = b.y;
        sW2t[(c4 + 2) * WSTR + k] = b.z;
        sW2t[(c4 + 3) * WSTR + k] = b.w;
    }
    if (t < GIN_D) { sB1[t] = b1[t]; sB2[t] = b2[t]; }
    __syncthreads();

    const int row0 = (blockIdx.x * 4 + wave) * 16;
    float* tile = sT[wave];

    // Coalesced load of A = hin + agg (16 rows x 16 float4) into LDS.
    for (int i = lane; i < 256; i += 32) {
        int r  = i >> 4;
        int c4 = (i & 15) << 2;
        int gr = row0 + r;
        float4 v = make_float4(0.f, 0.f, 0.f, 0.f);
        if (gr < n) {
            const float4 a = *(const float4*)(hin + (size_t)gr * GIN_D + c4);
            const float4 g = *(const float4*)(agg + (size_t)gr * GIN_D + c4);
            v.x = a.x + g.x; v.y = a.y + g.y; v.z = a.z + g.z; v.w = a.w + g.w;
        }
        tile[r * TSTR + c4 + 0] = v.x;
        tile[r * TSTR + c4 + 1] = v.y;
        tile[r * TSTR + c4 + 2] = v.z;
        tile[r * TSTR + c4 + 3] = v.w;
    }
    __syncthreads();

    // A fragments: 16x4 f32 layout -> lane m holds K = 4*kb + 2*hi (+1).
    v2f A[16];
#pragma unroll
    for (int kb = 0; kb < 16; ++kb) {
        int k = 4 * kb + 2 * hi;
        A[kb].x = tile[m * TSTR + k];
        A[kb].y = tile[m * TSTR + k + 1];
    }
    __syncthreads();

    // ---- Stage 1: T = relu(A @ W1 + b1), written back to tile ----
    for (int nt = 0; nt < 4; ++nt) {
        const int n0 = nt * 16;
        float bv = sB1[n0 + m];
        v8f acc = {bv, bv, bv, bv, bv, bv, bv, bv};
#pragma unroll
        for (int kb = 0; kb < 16; ++kb) {
            int k = 4 * kb + 2 * hi;
            v2f B;
            B.x = sW1t[(n0 + m) * WSTR + k];
            B.y = sW1t[(n0 + m) * WSTR + k + 1];
            acc = __builtin_amdgcn_wmma_f32_16x16x4_f32(false, A[kb], false, B,
                                                        (short)0, acc, false, false);
        }
#pragma unroll
        for (int j = 0; j < 8; ++j)
            tile[(j + 8 * hi) * TSTR + n0 + m] = fmaxf(acc[j], 0.f);  // C/D layout -> LDS
    }
    __syncthreads();

    // Reload T as A fragments for stage 2.
#pragma unroll
    for (int kb = 0; kb < 16; ++kb) {
        int k = 4 * kb + 2 * hi;
        A[kb].x = tile[m * TSTR + k];
        A[kb].y = tile[m * TSTR + k + 1];
    }

    // ---- Stage 2: OUT = relu(T @ W2 + b2) -> global ----
    for (int nt = 0; nt < 4; ++nt) {
        const int n0 = nt * 16;
        float bv = sB2[n0 + m];
        v8f acc = {bv, bv, bv, bv, bv, bv, bv, bv};
#pragma unroll
        for (int kb = 0; kb < 16; ++kb) {
            int k = 4 * kb + 2 * hi;
            v2f B;
            B.x = sW2t[(n0 + m) * WSTR + k];
            B.y = sW2t[(n0 + m) * WSTR + k + 1];
            acc = __builtin_amdgcn_wmma_f32_16x16x4_f32(false, A[kb], false, B,
                                                        (short)0, acc, false, false);
        }
#pragma unroll
        for (int j = 0; j < 8; ++j) {
            int gr = row0 + j + 8 * hi;
            if (gr < n) hout[(size_t)gr * GIN_D + n0 + m] = fmaxf(acc[j], 0.f);
        }
    }
}

// ---------------- readout: out[i] = dot(h[i], out_w) + out_b ----------------
__global__ __launch_bounds__(256) void gin_readout_kernel(const float* __restrict__ h,
                                                          const float* __restrict__ ow,
                                                          const float* __restrict__ ob,
                                                          float* __restrict__ out, int n) {
    int gw   = (blockIdx.x * 256 + threadIdx.x) >> 5;  // one wave per node
    int lane = threadIdx.x & 31;
    if (gw >= n) return;
    float s = h[(size_t)gw * GIN_D + lane] * ow[lane] +
              h[(size_t)gw * GIN_D + 32 + lane] * ow[32 + lane];
    for (int off = 16; off; off >>= 1) s += __shfl_xor(s, off, 32);
    if (lane == 0) out[gw] = s + ob[0];
}

extern "C" void kernel_launch(void* const* d_in, const int* in_sizes, int n_in,
                              void* d_out, int out_size, void* d_ws, size_t ws_size,
                              hipStream_t stream) {
    const float* x  = (const float*)d_in[0];
    const int*   ei = (const int*)d_in[1];
    const int N = in_sizes[0] / GIN_D;
    const int E = in_sizes[1] / 2;
    const int* src = ei;
    const int* dst = ei + E;
    const float* W1[3] = {(const float*)d_in[2], (const float*)d_in[6],  (const float*)d_in[10]};
    const float* B1[3] = {(const float*)d_in[3], (const float*)d_in[7],  (const float*)d_in[11]};
    const float* W2[3] = {(const float*)d_in[4], (const float*)d_in[8],  (const float*)d_in[12]};
    const float* B2[3] = {(const float*)d_in[5], (const float*)d_in[9],  (const float*)d_in[13]};
    const float* out_w = (const float*)d_in[14];
    const float* out_b = (const float*)d_in[15];
    float* out = (float*)d_out;

    float* agg = (float*)d_ws;
    float* hA  = agg + (size_t)N * GIN_D;
    float* hB  = hA  + (size_t)N * GIN_D;

    const int n4 = N * GIN_D / 4;
    dim3 zg((unsigned)((n4 + 255) / 256)), b256(256);
    dim3 ag((unsigned)(((long long)E * 16 + 255) / 256));
    dim3 mg((unsigned)((N + 63) / 64)), mb(128);
    dim3 rg((unsigned)(((long long)N * 32 + 255) / 256));

    const float* hin = x;
    float* bufs[2] = {hA, hB};
    for (int l = 0; l < 3; ++l) {
        gin_zero_kernel<<<zg, b256, 0, stream>>>((float4*)agg, n4);
        gin_agg_kernel<<<ag, b256, 0, stream>>>(hin, src, dst, agg, E);
        float* hout = bufs[l & 1];
        gin_mlp_kernel<<<mg, mb, 0, stream>>>(hin, agg, W1[l], B1[l], W2[l], B2[l], hout, N);
        hin = hout;
    }
    gin_readout_kernel<<<rg, b256, 0, stream>>>(hin, out_w, out_b, out, N);
}